// GATClassifier_22033182228597
// MI455X (gfx1250) — compile-verified
//
#include <hip/hip_runtime.h>
#include <math.h>

#define NEG_SLOPE 0.2f
#define IN_DIM    128
#define OUT_DIM   256   // H*F
#define H_DIM     4
#define F_DIM     64

typedef float    v2f  __attribute__((ext_vector_type(2)));
typedef float    v4f  __attribute__((ext_vector_type(4)));
typedef float    v8f  __attribute__((ext_vector_type(8)));

// native f32 atomic add (global_atomic_add_f32, no CAS loop)
__device__ __forceinline__ void atomAddF32(float* p, float v) {
    unsafeAtomicAdd(p, v);
}

// ---------------------------------------------------------------------------
// K1: feat [N,128] @ W.T [128,256] -> fhf [N,256] via WMMA
//     one wave computes a 16x64 tile: 1 A-fragment reused by 4 B-fragments
//     => 128 v_wmma_f32_16x16x4_f32 per wave, 4x less feat traffic
// ---------------------------------------------------------------------------
__global__ void gat_gemm_wmma(const float* __restrict__ A,   // feat
                              const float* __restrict__ Wm,  // W [256,128]
                              float* __restrict__ C,         // fhf [N,256]
                              int nRows) {
    const int lane = threadIdx.x & 31;
    const int wave = blockIdx.x * (blockDim.x >> 5) + (threadIdx.x >> 5);
    const int tilesG = OUT_DIM / 64;           // 4 column groups of 64
    const int tm = wave / tilesG;
    const int tg = wave % tilesG;
    if (tm * 16 >= nRows) return;              // wave-uniform exit

    const int row  = tm * 16 + (lane & 15);
    const int rowA = row < nRows ? row : nRows - 1;   // branch-free clamp
    const int col0 = tg * 64 + (lane & 15);
    const int khalf = (lane >> 4) * 2;         // lanes 16..31 hold K+2,K+3

    const float* __restrict__ arow = A  + (size_t)rowA * IN_DIM + khalf;
    const float* __restrict__ b0 = Wm + (size_t)(col0     ) * IN_DIM + khalf;
    const float* __restrict__ b1 = Wm + (size_t)(col0 + 16) * IN_DIM + khalf;
    const float* __restrict__ b2 = Wm + (size_t)(col0 + 32) * IN_DIM + khalf;
    const float* __restrict__ b3 = Wm + (size_t)(col0 + 48) * IN_DIM + khalf;

    v8f c0 = {}, c1 = {}, c2 = {}, c3 = {};
    #pragma unroll
    for (int k = 0; k < IN_DIM; k += 4) {
        v2f a, bb;
        a.x = arow[k];  a.y = arow[k + 1];
        bb.x = b0[k];   bb.y = b0[k + 1];
        c0 = __builtin_amdgcn_wmma_f32_16x16x4_f32(false, a, false, bb, (short)0, c0, false, false);
        bb.x = b1[k];   bb.y = b1[k + 1];
        c1 = __builtin_amdgcn_wmma_f32_16x16x4_f32(false, a, false, bb, (short)0, c1, false, false);
        bb.x = b2[k];   bb.y = b2[k + 1];
        c2 = __builtin_amdgcn_wmma_f32_16x16x4_f32(false, a, false, bb, (short)0, c2, false, false);
        bb.x = b3[k];   bb.y = b3[k + 1];
        c3 = __builtin_amdgcn_wmma_f32_16x16x4_f32(false, a, false, bb, (short)0, c3, false, false);
    }
    // C layout: VGPR i -> M = tm*16 + i + 8*(lane>=16), N = col group + (lane&15)
    const int mBase = tm * 16 + ((lane >> 4) << 3);
    #pragma unroll
    for (int i = 0; i < 8; ++i) {
        const int m = mBase + i;
        if (m < nRows) {
            float* __restrict__ crow = C + (size_t)m * OUT_DIM;
            crow[col0]      = c0[i];
            crow[col0 + 16] = c1[i];
            crow[col0 + 32] = c2[i];
            crow[col0 + 48] = c3[i];
        }
    }
}

// ---------------------------------------------------------------------------
// K2: el[n,h] = <fhf[n,h,:], attn_l[h,:]> ; er likewise
// ---------------------------------------------------------------------------
__global__ void gat_attn_proj(const float* __restrict__ fhf,
                              const float* __restrict__ al,
                              const float* __restrict__ ar,
                              float* __restrict__ el, float* __restrict__ er,
                              int N) {
    const int idx = blockIdx.x * blockDim.x + threadIdx.x;   // over N*H
    if (idx >= N * H_DIM) return;
    const int n = idx >> 2, h = idx & 3;
    const float* __restrict__ v = fhf + (size_t)n * OUT_DIM + h * F_DIM;
    float sl = 0.f, sr = 0.f;
    #pragma unroll 8
    for (int f = 0; f < F_DIM; ++f) {
        sl += v[f] * al[h * F_DIM + f];
        sr += v[f] * ar[h * F_DIM + f];
    }
    el[idx] = sl;
    er[idx] = sr;
}

// order-preserving float<->uint encode for atomic segment-max
__device__ __forceinline__ unsigned f2ord(float x) {
    unsigned u = __float_as_uint(x);
    return (u & 0x80000000u) ? ~u : (u | 0x80000000u);
}
__device__ __forceinline__ float ord2f(unsigned u) {
    return (u & 0x80000000u) ? __uint_as_float(u & 0x7FFFFFFFu)
                             : __uint_as_float(~u);
}

// ---------------------------------------------------------------------------
// K3: e = leaky_relu(el[src]+er[dst]); segment max into m[dst]
// ---------------------------------------------------------------------------
__global__ void gat_edge_logits(const float* __restrict__ el,
                                const float* __restrict__ er,
                                const int* __restrict__ src,
                                const int* __restrict__ dst,
                                float* __restrict__ ebuf,
                                unsigned* __restrict__ mbuf, int E) {
    const int idx = blockIdx.x * blockDim.x + threadIdx.x;   // over E*H
    if (idx >= E * H_DIM) return;
    const int ed = idx >> 2, h = idx & 3;
    const int s = src[ed], d = dst[ed];
    float x = el[s * H_DIM + h] + er[d * H_DIM + h];
    x = x > 0.f ? x : NEG_SLOPE * x;
    ebuf[idx] = x;
    atomicMax(&mbuf[d * H_DIM + h], f2ord(x));
}

// ---------------------------------------------------------------------------
// K4: ex = exp(e - m[dst]); segment sum into s[dst]; ex overwrites ebuf
// ---------------------------------------------------------------------------
__global__ void gat_edge_exp(float* __restrict__ ebuf,
                             const int* __restrict__ dst,
                             const unsigned* __restrict__ mbuf,
                             float* __restrict__ ssum, int E) {
    const int idx = blockIdx.x * blockDim.x + threadIdx.x;
    if (idx >= E * H_DIM) return;
    const int ed = idx >> 2, h = idx & 3;
    const int d = dst[ed];
    const float ex = __expf(ebuf[idx] - ord2f(mbuf[d * H_DIM + h]));
    ebuf[idx] = ex;
    atomAddF32(&ssum[d * H_DIM + h], ex);
}

// ---------------------------------------------------------------------------
// K5: a = ex/s[dst]; scatter-add a*fhf[src] into rst and a*feat[src] into agg
//     one 128-thread block per edge; coalesced reads, native L2 f32 atomics
// ---------------------------------------------------------------------------
__global__ void gat_edge_aggregate(const float* __restrict__ feat,
                                   const float* __restrict__ fhf,
                                   const float* __restrict__ ebuf,
                                   const float* __restrict__ ssum,
                                   const int* __restrict__ src,
                                   const int* __restrict__ dst,
                                   float* __restrict__ rst,   // [N,H,F] in d_out
                                   float* __restrict__ agg,   // [N,H,IN]
                                   int E) {
    const int ed = blockIdx.x;
    const int t  = threadIdx.x;                 // 0..127
    const int s = src[ed], d = dst[ed];
    const float fv = feat[(size_t)s * IN_DIM + t];
    #pragma unroll
    for (int h = 0; h < H_DIM; ++h) {
        const float a = ebuf[ed * H_DIM + h] / ssum[d * H_DIM + h];
        atomAddF32(&agg[((size_t)d * H_DIM + h) * IN_DIM + t], fv * a);
        if (t < F_DIM)
            atomAddF32(&rst[((size_t)d * H_DIM + h) * F_DIM + t],
                       fhf[(size_t)s * OUT_DIM + h * F_DIM + t] * a);
    }
}

// ---------------------------------------------------------------------------
// K6: ReLU in place on rst region of d_out
// ---------------------------------------------------------------------------
__global__ void gat_relu_inplace(float* __restrict__ p, int n) {
    const int i = blockIdx.x * blockDim.x + threadIdx.x;
    if (i < n) p[i] = fmaxf(p[i], 0.f);
}

// ---------------------------------------------------------------------------
// K7: out_edge[e,:,:] = |agg[src[e],:,:] - agg[dst[e],:,:]|   (512 floats/edge)
//     one block per edge, 128 threads x float4 => b128 loads/stores.
//     Gathers of agg (51 MB) stay L2-resident; the 819 MB output stream is
//     written with non-temporal hint so it doesn't evict agg from L2.
// ---------------------------------------------------------------------------
__global__ void gat_edge_out(const v4f* __restrict__ agg4,  // [N, 128] v4f
                             const int* __restrict__ src,
                             const int* __restrict__ dst,
                             v4f* __restrict__ out4, int E) {
    const int ed = blockIdx.x;
    const int t  = threadIdx.x;                 // 0..127 -> H*IN/4 chunks
    const int s = src[ed], d = dst[ed];
    const v4f a = agg4[(size_t)s * (H_DIM * IN_DIM / 4) + t];
    const v4f b = agg4[(size_t)d * (H_DIM * IN_DIM / 4) + t];
    v4f r;
    r.x = fabsf(a.x - b.x);
    r.y = fabsf(a.y - b.y);
    r.z = fabsf(a.z - b.z);
    r.w = fabsf(a.w - b.w);
    __builtin_nontemporal_store(r, &out4[(size_t)ed * (H_DIM * IN_DIM / 4) + t]);
}

// ---------------------------------------------------------------------------
extern "C" void kernel_launch(void* const* d_in, const int* in_sizes, int n_in,
                              void* d_out, int out_size, void* d_ws, size_t ws_size,
                              hipStream_t stream) {
    const float* feat   = (const float*)d_in[0];
    const float* W      = (const float*)d_in[1];
    const float* attn_l = (const float*)d_in[2];
    const float* attn_r = (const float*)d_in[3];
    const int*   src    = (const int*)d_in[4];
    const int*   dst    = (const int*)d_in[5];

    const int HF = in_sizes[2];          // 256
    const int IN = in_sizes[1] / HF;     // 128
    const int N  = in_sizes[0] / IN;     // 25000
    const int E  = in_sizes[4];          // 400000

    float* out_rst  = (float*)d_out;                       // N*H*F
    float* out_edge = (float*)d_out + (size_t)N * HF;      // E*H*IN

    // workspace layout (floats)
    float* ws   = (float*)d_ws;
    size_t o    = 0;
    float*    fhf  = ws + o;  o += (size_t)N * HF;         // 6.4M
    float*    el   = ws + o;  o += (size_t)N * H_DIM;
    float*    er   = ws + o;  o += (size_t)N * H_DIM;
    float*    ebuf = ws + o;  o += (size_t)E * H_DIM;      // 1.6M
    unsigned* mbuf = (unsigned*)(ws + o); o += (size_t)N * H_DIM;
    float*    ssum = ws + o;  o += (size_t)N * H_DIM;
    float*    agg  = ws + o;  o += (size_t)N * H_DIM * IN; // 12.8M

    // zero-init accumulators (capturable async memsets)
    hipMemsetAsync(out_rst, 0, (size_t)N * HF * sizeof(float), stream);
    hipMemsetAsync(mbuf,    0, (size_t)N * H_DIM * sizeof(unsigned), stream);
    hipMemsetAsync(ssum,    0, (size_t)N * H_DIM * sizeof(float), stream);
    hipMemsetAsync(agg,     0, (size_t)N * H_DIM * IN * sizeof(float), stream);

    // K1: WMMA GEMM — one wave per 16x64 tile, 4 waves (128 threads) / block
    const int tilesM = (N + 15) / 16;
    const int tilesG = HF / 64;
    const int totalWaves = tilesM * tilesG;
    const int wavesPerBlock = 4;
    gat_gemm_wmma<<<(totalWaves + wavesPerBlock - 1) / wavesPerBlock,
                    wavesPerBlock * 32, 0, stream>>>(feat, W, fhf, N);

    // K2: attention projections
    gat_attn_proj<<<(N * H_DIM + 255) / 256, 256, 0, stream>>>(
        fhf, attn_l, attn_r, el, er, N);

    // K3: edge logits + segment max
    gat_edge_logits<<<(E * H_DIM + 255) / 256, 256, 0, stream>>>(
        el, er, src, dst, ebuf, mbuf, E);

    // K4: exp + segment sum
    gat_edge_exp<<<(E * H_DIM + 255) / 256, 256, 0, stream>>>(
        ebuf, dst, mbuf, ssum, E);

    // K5: normalize + scatter aggregation (rst into d_out, agg_ori into ws)
    gat_edge_aggregate<<<E, IN, 0, stream>>>(
        feat, fhf, ebuf, ssum, src, dst, out_rst, agg, E);

    // K6: ReLU on rst
    gat_relu_inplace<<<(N * HF + 255) / 256, 256, 0, stream>>>(out_rst, N * HF);

    // K7: |agg[src] - agg[dst]| streamed output (819 MB, HBM-bound, NT stores)
    gat_edge_out<<<E, IN, 0, stream>>>((const v4f*)agg, src, dst,
                                       (v4f*)out_edge, E);
}